// PtModule_76166950027718
// MI455X (gfx1250) — compile-verified
//
#include <hip/hip_runtime.h>

// Elementwise y = (x + 2) * 1.5 over 8192*8192 fp32 — pure HBM stream
// (512 MiB total @ 23.3 TB/s => ~23 us floor).
//
// CDNA5 path: 4-deep software pipeline of GLOBAL_LOAD_ASYNC_TO_LDS_B128
// (ASYNCcnt, in-order completion, th:TH_LOAD_NT), ds_load_b128 consume,
// non-temporal global_store_b128 produce. 4 outstanding async loads/wave
// = 2 KB/wave in flight; 16 KB LDS per 256-thread block (of 320 KB/WGP).
// Host precomputes stride/iters/pchunks so no device-side division.

typedef float v4f __attribute__((ext_vector_type(4)));

#define DEPTH 4u
#define TPB   256u

__global__ __launch_bounds__(TPB) void ew_async_kernel(const float* __restrict__ src,
                                                       float* __restrict__ dst,
                                                       unsigned n,
                                                       unsigned stride,   // total threads
                                                       unsigned iters,    // nvec / stride
                                                       unsigned pchunks)  // pipelined chunk count
{
    __shared__ v4f smem[DEPTH][TPB];

    const unsigned t    = threadIdx.x;
    const unsigned nvec = n >> 2;                        // float4 chunks
    const unsigned gtid = blockIdx.x * TPB + t;

    const v4f* __restrict__ srcv = (const v4f*)src;
    v4f* __restrict__ dstv       = (v4f*)dst;

    // LDS byte offsets of this lane's 16B slot in each ring-buffer stage
    // (low 32 bits of the generic shared-aperture address == LDS offset).
    unsigned lds[DEPTH];
#pragma unroll
    for (unsigned b = 0; b < DEPTH; ++b)
        lds[b] = (unsigned)(unsigned long long)&smem[b][t];

    // ---- pipelined uniform region: chunks [0, iters*stride), iters >= DEPTH ----
    if (pchunks) {
        // Prologue: fill the pipeline with DEPTH async copies (NT: zero reuse).
#pragma unroll
        for (unsigned b = 0; b < DEPTH; ++b) {
            const v4f* g = srcv + (gtid + b * stride);
            asm volatile("global_load_async_to_lds_b128 %0, %1, off th:TH_LOAD_NT"
                         :: "v"(lds[b]), "v"(g) : "memory");
        }

        // Steady state: wait-oldest, consume, store, reissue into freed slot.
        // The store's implicit DScnt wait + program order (volatile asm with
        // memory clobber) make the LDS slot reuse WAR-safe.
        unsigned k = 0;
        for (; k + DEPTH < iters; ++k) {
            asm volatile("s_wait_asynccnt 3" ::: "memory");   // oldest of 4 landed
            v4f x = smem[k & 3u][t];                          // ds_load_b128
            v4f y = (x + 2.0f) * 1.5f;                        // (x+2)*3/2 fused
            __builtin_nontemporal_store(y, dstv + (gtid + k * stride));
            const v4f* g = srcv + (gtid + (k + DEPTH) * stride);
            asm volatile("global_load_async_to_lds_b128 %0, %1, off th:TH_LOAD_NT"
                         :: "v"(lds[k & 3u]), "v"(g) : "memory");
        }

        // Drain: remaining DEPTH loads were issued long ago; wait once, consume.
        asm volatile("s_wait_asynccnt 0" ::: "memory");
        for (; k < iters; ++k) {
            v4f x = smem[k & 3u][t];
            v4f y = (x + 2.0f) * 1.5f;
            __builtin_nontemporal_store(y, dstv + (gtid + k * stride));
        }
    }

    // ---- generic remainder: direct NT load/compute/store (not hit for 8192^2
    //      with the launch below, kept for correctness on any size) ----
    for (unsigned i = pchunks + gtid; i < nvec; i += stride) {
        v4f x = __builtin_nontemporal_load(srcv + i);
        v4f y = (x + 2.0f) * 1.5f;
        __builtin_nontemporal_store(y, dstv + i);
    }

    // ---- scalar tail for n % 4 != 0 ----
    const unsigned tail = n & 3u;
    if (blockIdx.x == 0 && t < tail) {
        unsigned idx = (nvec << 2) + t;
        dst[idx] = (src[idx] + 2.0f) * 1.5f;
    }
}

extern "C" void kernel_launch(void* const* d_in, const int* in_sizes, int n_in,
                              void* d_out, int out_size, void* d_ws, size_t ws_size,
                              hipStream_t stream) {
    const float* src = (const float*)d_in[0];
    float* dst       = (float*)d_out;
    const unsigned n = (unsigned)in_sizes[0];     // 67,108,864

    unsigned nvec   = n >> 2;                      // 2^24 float4 chunks
    unsigned blocks = (nvec + TPB - 1) / TPB;
    if (blocks > 8192u) blocks = 8192u;            // stride 2^21 -> 8 iters/thread
    if (blocks == 0u)   blocks = 1u;

    const unsigned stride  = blocks * TPB;
    const unsigned iters   = nvec / stride;                       // uniform trips
    const unsigned pchunks = (iters >= DEPTH) ? iters * stride : 0u;

    ew_async_kernel<<<dim3(blocks), dim3(TPB), 0, stream>>>(src, dst, n,
                                                            stride, iters, pchunks);
}